// SVDLayer_22668837388536
// MI455X (gfx1250) — compile-verified
//
#include <hip/hip_runtime.h>
#include <stdint.h>

// Batched 3x3 polar decomposition: R = U * V^T from SVD(A).
// Newton iteration with Frobenius scaling: X <- 0.5*(mu*X + (1/mu)*X^-T),
// X^-T = cofactor(X)/det(X). Converges to the orthogonal polar factor,
// which equals U*V^T exactly (sign of det preserved).
//
// Data path (CDNA5): wave-private async DMA staging.
//   Each wave copies its 32 matrices (288 dwords) global->LDS with
//   global_load_async_to_lds_b32 (lane-linear, fully coalesced, NT hint),
//   waits on ASYNCcnt, computes in f32 VALU, writes results to LDS, then
//   streams LDS->global with global_store_async_from_lds_b32 (NT hint).

#define MATS_PER_BLOCK 256
#define DW_PER_BLOCK   (MATS_PER_BLOCK * 9)   // 2304 dwords = 9216 B LDS
#define NITER          6

__global__ __launch_bounds__(256) void polar3x3_kernel(
    const float* __restrict__ in, float* __restrict__ out) {
  __shared__ float smem[DW_PER_BLOCK];

  const int tid  = threadIdx.x;
  const int lane = tid & 31;
  const int wave = tid >> 5;

  // ---- Stage in: this wave's 288 dwords, lane-linear (coalesced) ----------
  const uint32_t dword   = (uint32_t)wave * 288u + (uint32_t)lane;
  const uint32_t byteoff = dword * 4u;                 // global VADDR (32-bit off)
  // low 32 bits of a generic LDS pointer == LDS byte offset on AMDGCN
  const uint32_t ldsaddr = (uint32_t)(uintptr_t)smem + byteoff;

  const float* gbase = in + (size_t)blockIdx.x * DW_PER_BLOCK;

  // NOTE: INST_OFFSET is applied to BOTH the LDS and global address per the
  // CDNA5 async pseudocode, so one (vdst, vaddr) pair covers all 9 copies.
  asm volatile(
      "global_load_async_to_lds_b32 %0, %1, %2 offset:0 th:TH_LOAD_NT\n\t"
      "global_load_async_to_lds_b32 %0, %1, %2 offset:128 th:TH_LOAD_NT\n\t"
      "global_load_async_to_lds_b32 %0, %1, %2 offset:256 th:TH_LOAD_NT\n\t"
      "global_load_async_to_lds_b32 %0, %1, %2 offset:384 th:TH_LOAD_NT\n\t"
      "global_load_async_to_lds_b32 %0, %1, %2 offset:512 th:TH_LOAD_NT\n\t"
      "global_load_async_to_lds_b32 %0, %1, %2 offset:640 th:TH_LOAD_NT\n\t"
      "global_load_async_to_lds_b32 %0, %1, %2 offset:768 th:TH_LOAD_NT\n\t"
      "global_load_async_to_lds_b32 %0, %1, %2 offset:896 th:TH_LOAD_NT\n\t"
      "global_load_async_to_lds_b32 %0, %1, %2 offset:1024 th:TH_LOAD_NT\n\t"
      "s_wait_asynccnt 0"
      :
      : "v"(ldsaddr), "v"(byteoff), "s"(gbase)
      : "memory");

  // ---- Load my 3x3 matrix from LDS (stride-9 banks: conflict-free) --------
  const int m = tid * 9;
  float x00 = smem[m + 0], x01 = smem[m + 1], x02 = smem[m + 2];
  float x10 = smem[m + 3], x11 = smem[m + 4], x12 = smem[m + 5];
  float x20 = smem[m + 6], x21 = smem[m + 7], x22 = smem[m + 8];

  // ---- Scaled Newton polar iteration --------------------------------------
#pragma unroll
  for (int it = 0; it < NITER; ++it) {
    // Cofactor matrix C: X^{-T} = C / det(X)
    const float c00 = x11 * x22 - x12 * x21;
    const float c01 = x12 * x20 - x10 * x22;
    const float c02 = x10 * x21 - x11 * x20;
    const float c10 = x02 * x21 - x01 * x22;
    const float c11 = x00 * x22 - x02 * x20;
    const float c12 = x01 * x20 - x00 * x21;
    const float c20 = x01 * x12 - x02 * x11;
    const float c21 = x02 * x10 - x00 * x12;
    const float c22 = x00 * x11 - x01 * x10;

    const float det = x00 * c00 + x01 * c01 + x02 * c02;
    // ~1 ulp v_rcp_f32 is plenty: error is absorbed by the quadratically
    // convergent iteration, and det -> +-1 at convergence.
    const float ad     = fmaxf(fabsf(det), 1e-30f);
    const float invdet = __builtin_copysignf(__builtin_amdgcn_rcpf(ad), det);

    const float y00 = c00 * invdet, y01 = c01 * invdet, y02 = c02 * invdet;
    const float y10 = c10 * invdet, y11 = c11 * invdet, y12 = c12 * invdet;
    const float y20 = c20 * invdet, y21 = c21 * invdet, y22 = c22 * invdet;

    // Frobenius scaling: mu = (||Y||_F / ||X||_F)^(1/2); fold the 0.5 in.
    const float nx = x00 * x00 + x01 * x01 + x02 * x02 +
                     x10 * x10 + x11 * x11 + x12 * x12 +
                     x20 * x20 + x21 * x21 + x22 * x22;
    const float ny = y00 * y00 + y01 * y01 + y02 * y02 +
                     y10 * y10 + y11 * y11 + y12 * y12 +
                     y20 * y20 + y21 * y21 + y22 * y22;
    const float q   = __builtin_amdgcn_sqrtf(nx * __builtin_amdgcn_rcpf(ny)); // (nx/ny)^.5
    const float rmu = __builtin_amdgcn_sqrtf(q);                              // 1/mu
    const float hm  = 0.5f * __builtin_amdgcn_rcpf(rmu);                      // 0.5*mu
    const float hr  = 0.5f * rmu;                                             // 0.5/mu

    x00 = hm * x00 + hr * y00;
    x01 = hm * x01 + hr * y01;
    x02 = hm * x02 + hr * y02;
    x10 = hm * x10 + hr * y10;
    x11 = hm * x11 + hr * y11;
    x12 = hm * x12 + hr * y12;
    x20 = hm * x20 + hr * y20;
    x21 = hm * x21 + hr * y21;
    x22 = hm * x22 + hr * y22;
  }

  // ---- Stage out: LDS, then async LDS -> global ---------------------------
  smem[m + 0] = x00; smem[m + 1] = x01; smem[m + 2] = x02;
  smem[m + 3] = x10; smem[m + 4] = x11; smem[m + 5] = x12;
  smem[m + 6] = x20; smem[m + 7] = x21; smem[m + 8] = x22;

  float* obase = out + (size_t)blockIdx.x * DW_PER_BLOCK;
  asm volatile(
      "s_wait_dscnt 0\n\t"  // my ds_stores complete before DMA reads LDS
      "global_store_async_from_lds_b32 %1, %0, %2 offset:0 th:TH_STORE_NT\n\t"
      "global_store_async_from_lds_b32 %1, %0, %2 offset:128 th:TH_STORE_NT\n\t"
      "global_store_async_from_lds_b32 %1, %0, %2 offset:256 th:TH_STORE_NT\n\t"
      "global_store_async_from_lds_b32 %1, %0, %2 offset:384 th:TH_STORE_NT\n\t"
      "global_store_async_from_lds_b32 %1, %0, %2 offset:512 th:TH_STORE_NT\n\t"
      "global_store_async_from_lds_b32 %1, %0, %2 offset:640 th:TH_STORE_NT\n\t"
      "global_store_async_from_lds_b32 %1, %0, %2 offset:768 th:TH_STORE_NT\n\t"
      "global_store_async_from_lds_b32 %1, %0, %2 offset:896 th:TH_STORE_NT\n\t"
      "global_store_async_from_lds_b32 %1, %0, %2 offset:1024 th:TH_STORE_NT\n\t"
      "s_wait_asynccnt 0"
      :
      : "v"(ldsaddr), "v"(byteoff), "s"(obase)
      : "memory");
}

extern "C" void kernel_launch(void* const* d_in, const int* in_sizes, int n_in,
                              void* d_out, int out_size, void* d_ws, size_t ws_size,
                              hipStream_t stream) {
  (void)n_in; (void)out_size; (void)d_ws; (void)ws_size;
  const float* in  = (const float*)d_in[0];
  float*       out = (float*)d_out;
  const int batch  = in_sizes[0] / 9;          // 2,097,152 (divisible by 256)
  const int blocks = batch / MATS_PER_BLOCK;   // 8192
  polar3x3_kernel<<<blocks, MATS_PER_BLOCK, 0, stream>>>(in, out);
}